// ResidualVQ_4947802325585
// MI455X (gfx1250) — compile-verified
//
#include <hip/hip_runtime.h>
#include <hip/hip_bf16.h>

typedef float v2f __attribute__((ext_vector_type(2)));
typedef float v8f __attribute__((ext_vector_type(8)));

#define R_STAGES 8
#define K_CODES 1024
#define DIM 256
#define B_ROWS 65536

#define M_TILE 64        // rows of z per workgroup
#define N_CHUNK 64       // codebook columns staged in LDS at a time
#define THREADS 128      // 4 waves of 32
#define RES_STRIDE 260   // padded LDS row stride (floats) -> conflict-free strided reads
#define CB_STRIDE 260

// ---------------------------------------------------------------------------
// Zero the loss accumulator (graph-replay safe: runs every launch).
// ---------------------------------------------------------------------------
__global__ void rvq_zero_kernel(float* loss) { *loss = 0.0f; }

// ---------------------------------------------------------------------------
// Precompute ||c_k||^2 for all R*K codewords into workspace.
// ---------------------------------------------------------------------------
__global__ void rvq_cnorm_kernel(const float* __restrict__ cb,
                                 float* __restrict__ cnorm) {
  int e = blockIdx.x * blockDim.x + threadIdx.x;      // 0 .. R*K-1
  const float4* p = (const float4*)(cb + (size_t)e * DIM);
  float s = 0.f;
#pragma unroll 8
  for (int i = 0; i < DIM / 4; ++i) {
    float4 v = p[i];
    s += v.x * v.x + v.y * v.y + v.z * v.z + v.w * v.w;
  }
  cnorm[e] = s;
}

// ---------------------------------------------------------------------------
// Main fused residual-VQ kernel.
//   - residual tile (64 x 256 f32) resident in LDS across all 8 stages
//   - per stage: stream codebook in 64-col chunks to LDS, score with
//     v_wmma_f32_16x16x4_f32, track per-row argmin, gather+subtract.
// ---------------------------------------------------------------------------
__launch_bounds__(THREADS)
__global__ void rvq_main_kernel(const float* __restrict__ z,
                                const float* __restrict__ cb,
                                const float* __restrict__ cnorm,
                                float* __restrict__ out_z,
                                int* __restrict__ out_idx,
                                float* __restrict__ out_loss) {
  extern __shared__ char smem[];
  float* lds_res = (float*)smem;                       // 64*260 floats
  float* lds_cb  = lds_res + M_TILE * RES_STRIDE;      // 64*260 floats
  float* lds_cn  = lds_cb + N_CHUNK * CB_STRIDE;       // 1024 floats
  int*   lds_idx = (int*)(lds_cn + K_CODES);           // 64 ints
  float* lds_red = (float*)(lds_idx + M_TILE);         // 1 float

  const int tid  = threadIdx.x;
  const int lane = tid & 31;
  const int wave = tid >> 5;
  const int half = lane >> 4;          // 0: K=0,1 / M-low half ; 1: K=2,3
  const int lh   = lane & 15;
  const int m0   = wave * 16;          // each wave owns a 16-row stripe
  const size_t row0 = (size_t)blockIdx.x * M_TILE;

  // ---- Load residual tile (= z) into LDS; also emit z_q_st = z copy. ----
  const float4* zp = (const float4*)(z + row0 * DIM);
  float4* op = (float4*)(out_z + row0 * DIM);
#pragma unroll
  for (int i = 0; i < (M_TILE * DIM / 4) / THREADS; ++i) {
    int e4 = tid + i * THREADS;              // row-major float4 index
    float4 v = zp[e4];
    op[e4] = v;                              // straight-through output == z
    int row = e4 >> 6, d4 = e4 & 63;
    *(float4*)&lds_res[row * RES_STRIDE + d4 * 4] = v;
  }

  for (int r = 0; r < R_STAGES; ++r) {
    const float* cbr = cb + (size_t)r * K_CODES * DIM;

    // stage ||c||^2 table into LDS
    for (int i = tid; i < K_CODES; i += THREADS)
      lds_cn[i] = cnorm[r * K_CODES + i];

    float minval[8];
    int   minidx[8];
#pragma unroll
    for (int v = 0; v < 8; ++v) { minval[v] = 3.4e38f; minidx[v] = 0; }

    for (int chunk = 0; chunk < K_CODES / N_CHUNK; ++chunk) {
      __syncthreads();  // protect lds_cb reuse + lds_res/lds_cn visibility
      // ---- cooperative stage of 64 codebook columns into LDS ----
      const float4* cp = (const float4*)(cbr + (size_t)chunk * N_CHUNK * DIM);
#pragma unroll
      for (int i = 0; i < (N_CHUNK * DIM / 4) / THREADS; ++i) {
        int e4 = tid + i * THREADS;
        float4 v = cp[e4];
        int col = e4 >> 6, d4 = e4 & 63;
        *(float4*)&lds_cb[col * CB_STRIDE + d4 * 4] = v;
      }
      if (chunk + 1 < K_CODES / N_CHUNK) {   // hint next chunk into caches
        __builtin_prefetch(cbr + (size_t)(chunk + 1) * N_CHUNK * DIM + tid * 16, 0, 0);
      }
      __syncthreads();

      // ---- 16x16 score tiles: D-accumulate with f32 WMMA ----
      v8f acc[4];
#pragma unroll
      for (int nt = 0; nt < 4; ++nt) acc[nt] = (v8f)0.0f;

#pragma unroll 4
      for (int kk = 0; kk < DIM / 4; ++kk) {
        const int d0 = kk * 4 + half * 2;    // lanes 0-15: K=0,1 ; 16-31: K=2,3
        v2f a = *(const v2f*)&lds_res[(m0 + lh) * RES_STRIDE + d0];
#pragma unroll
        for (int nt = 0; nt < 4; ++nt) {
          v2f b = *(const v2f*)&lds_cb[(nt * 16 + lh) * CB_STRIDE + d0];
          acc[nt] = __builtin_amdgcn_wmma_f32_16x16x4_f32(
              /*neg_a=*/false, a, /*neg_b=*/false, b,
              /*c_mod=*/(short)0, acc[nt], /*reuse_a=*/false, /*reuse_b=*/false);
        }
      }

      // ---- dist = ||c||^2 - 2*dot ; update per-row running argmin ----
#pragma unroll
      for (int nt = 0; nt < 4; ++nt) {
        int col = chunk * N_CHUNK + nt * 16 + lh;
        float cn = lds_cn[col];
#pragma unroll
        for (int v = 0; v < 8; ++v) {
          float dist = cn - 2.0f * acc[nt][v];
          if (dist < minval[v]) { minval[v] = dist; minidx[v] = col; }
        }
      }
    }  // chunks

    // ---- cross-lane argmin within each 16-lane group (first-index ties) ----
#pragma unroll
    for (int v = 0; v < 8; ++v) {
      float bv = minval[v];
      int   bi = minidx[v];
#pragma unroll
      for (int m = 8; m >= 1; m >>= 1) {
        float ov = __shfl_xor(bv, m, 32);
        int   oi = __shfl_xor(bi, m, 32);
        if (ov < bv || (ov == bv && oi < bi)) { bv = ov; bi = oi; }
      }
      if (lh == 0) {
        int row = m0 + v + 8 * half;         // C/D layout: lanes16-31 hold M+8
        lds_idx[row] = bi;
        out_idx[(row0 + row) * R_STAGES + r] = bi;  // indices[b][r]
      }
    }
    __syncthreads();

    // ---- residual update: res -= codebook[idx] (gathered from global) ----
#pragma unroll 4
    for (int i = 0; i < (M_TILE * DIM) / THREADS; ++i) {
      int e = tid + i * THREADS;
      int row = e >> 8, d = e & 255;
      int ksel = lds_idx[row];
      lds_res[row * RES_STRIDE + d] -= cbr[(size_t)ksel * DIM + d];
    }
    // next stage's chunk-0 barriers order these writes vs. WMMA reads
  }

  // ---- commitment loss = mean(res_final^2) ----
  __syncthreads();
  if (tid == 0) lds_red[0] = 0.0f;
  float s = 0.f;
#pragma unroll 4
  for (int i = 0; i < (M_TILE * DIM) / THREADS; ++i) {
    int e = tid + i * THREADS;
    float x = lds_res[(e >> 8) * RES_STRIDE + (e & 255)];
    s += x * x;
  }
#pragma unroll
  for (int m = 16; m >= 1; m >>= 1) s += __shfl_xor(s, m, 32);
  __syncthreads();
  if (lane == 0) atomicAdd(lds_red, s);
  __syncthreads();
  if (tid == 0)
    atomicAdd(out_loss, lds_red[0] * (1.0f / ((float)B_ROWS * (float)DIM)));
}

// ---------------------------------------------------------------------------
extern "C" void kernel_launch(void* const* d_in, const int* in_sizes, int n_in,
                              void* d_out, int out_size, void* d_ws, size_t ws_size,
                              hipStream_t stream) {
  const float* z  = (const float*)d_in[0];   // (B, D) f32
  const float* cb = (const float*)d_in[1];   // (R, K, D) f32

  float* out      = (float*)d_out;
  float* out_z    = out;                                            // (B, D)
  int*   out_idx  = (int*)(out + (size_t)B_ROWS * DIM);             // (B, R) int32
  float* out_loss = out + (size_t)B_ROWS * DIM + (size_t)B_ROWS * R_STAGES;  // scalar

  float* cnorm = (float*)d_ws;               // R*K floats = 32 KB scratch

  rvq_zero_kernel<<<1, 1, 0, stream>>>(out_loss);
  rvq_cnorm_kernel<<<(R_STAGES * K_CODES) / 256, 256, 0, stream>>>(cb, cnorm);

  size_t smem = (size_t)(M_TILE * RES_STRIDE + N_CHUNK * CB_STRIDE + K_CODES) * sizeof(float)
              + (size_t)M_TILE * sizeof(int) + 16;
  (void)hipFuncSetAttribute((const void*)rvq_main_kernel,
                            hipFuncAttributeMaxDynamicSharedMemorySize, (int)smem);
  rvq_main_kernel<<<B_ROWS / M_TILE, THREADS, smem, stream>>>(
      z, cb, cnorm, out_z, out_idx, out_loss);
}